// ViewControlPreprocessor_39247411151101
// MI455X (gfx1250) — compile-verified
//
#include <hip/hip_runtime.h>
#include <stdint.h>

// Problem constants (from reference setup_inputs; is_causal == False path):
//   hidden: (1536*1, 21, 3072) f32   -> rows = 32256, 768 float4/row
//   vcc:    (1, 81, 16) f32
//   out:    (1536, 21, 3200) f32     -> 800 float4/row (768 hidden + 32 tail)
#define N_ROWS   32256      // (B*S) * T_q = 1536 * 21
#define T_Q      21
#define C_HID4   768        // hidden float4 per row (= 3 chunks of 256)
#define C_OUT4   800
#define TAIL4    32         // group float4 per row (128 f32)
#define CHUNKS_PER_ROW 3
#define NBUF     4          // chunks staged in LDS per block iteration

// ---- device-side feature detection (evaluated per compilation pass; on the
// ---- host pass this is 0 and the fallback body is syntax-checked only) ----
#if defined(__has_builtin)
#  if __has_builtin(__builtin_amdgcn_global_load_async_to_lds_b128) && \
      __has_builtin(__builtin_amdgcn_global_store_async_from_lds_b128)
#    define DEV_HAVE_ASYNC 1
#  endif
#endif
#ifndef DEV_HAVE_ASYNC
#  define DEV_HAVE_ASYNC 0
#endif

// Exact parameter types expected by the async builtins (per hipcc diagnostics):
// both sides are pointers to 16-byte int vectors; global side in addrspace(1),
// LDS side in addrspace(3). vint4 is a native clang vector type (also valid
// for the nontemporal builtins, unlike HIP's float4 class).
typedef int vint4 __attribute__((vector_size(4 * sizeof(int))));
#define AS_GLOBAL_V4(p) ((__attribute__((address_space(1))) vint4*)(p))
#define AS_LDS_V4(p)    ((__attribute__((address_space(3))) vint4*)(p))

__device__ __forceinline__ void wait_async0() {
#if defined(__has_builtin) && __has_builtin(__builtin_amdgcn_s_wait_asynccnt)
    __builtin_amdgcn_s_wait_asynccnt(0);
#elif defined(__AMDGCN__)
    asm volatile("s_wait_asynccnt 0" ::: "memory");
#endif
}

// Bulk copy of hidden_states into the first 3072 f32 of each 3200-f32 output
// row. Chunk c (256 float4 = 4KB) comes from src[c*256 ..] and goes to
// dst[row*800 + part*256 ..] with row = c/3, part = c%3. Both sides are
// contiguous & 16B aligned -> pure B128 streams, global->LDS->global with no
// VGPR data movement (ASYNCcnt-tracked DMA path).
__global__ void __launch_bounds__(256)
bulk_copy_async(const vint4* __restrict__ src, vint4* __restrict__ dst,
                int nChunks) {
    __shared__ vint4 stage[NBUF][256];
    const int tid = threadIdx.x;
    for (int base = blockIdx.x * NBUF; base < nChunks; base += gridDim.x * NBUF) {
#if DEV_HAVE_ASYNC
        // Make sure stores consuming this LDS from a previous iteration drained.
        wait_async0();
#pragma unroll
        for (int k = 0; k < NBUF; ++k) {
            const int c = base + k;                 // wave-uniform predicate
            if (c < nChunks) {
                const vint4* g = src + (size_t)c * 256 + tid;
                __builtin_amdgcn_global_load_async_to_lds_b128(
                    AS_GLOBAL_V4(g), AS_LDS_V4(&stage[k][tid]), 0, 0);
            }
        }
        wait_async0();                              // all loads landed in LDS
#pragma unroll
        for (int k = 0; k < NBUF; ++k) {
            const int c = base + k;
            if (c < nChunks) {
                const int row  = c / CHUNKS_PER_ROW;
                const int part = c % CHUNKS_PER_ROW;
                vint4* g = dst + (size_t)row * C_OUT4 + part * 256 + tid;
                __builtin_amdgcn_global_store_async_from_lds_b128(
                    AS_GLOBAL_V4(g), AS_LDS_V4(&stage[k][tid]), 0, 0);
            }
        }
#else
        // Fallback: plain wide nontemporal streaming copy (still roofline-opt).
#pragma unroll
        for (int k = 0; k < NBUF; ++k) {
            const int c = base + k;
            if (c < nChunks) {
                const int row  = c / CHUNKS_PER_ROW;
                const int part = c % CHUNKS_PER_ROW;
                vint4 v = __builtin_nontemporal_load(src + (size_t)c * 256 + tid);
                __builtin_nontemporal_store(
                    v, dst + (size_t)row * C_OUT4 + part * 256 + tid);
            }
        }
#endif
    }
#if DEV_HAVE_ASYNC
    wait_async0();   // drain before wave retires
#endif
}

// Tail: out[row, 3072 + c] = vcc[max(4t + c/16 - 8, 0), c%16], t = row % 21.
// One float4 (4 channels, same frame since 4i never crosses a 16 boundary)
// per thread. Reads are a ~5KB working set -> L2/L0 resident.
__global__ void __launch_bounds__(256)
tail_group(const float* __restrict__ vcc, vint4* __restrict__ out4, int total) {
    const int idx = blockIdx.x * blockDim.x + threadIdx.x;
    if (idx >= total) return;
    const int row = idx / TAIL4;          // s*21 + t
    const int i   = idx % TAIL4;          // 0..31  (channels 4i..4i+3)
    const int t   = row % T_Q;
    const int j   = i >> 2;               // 0..7
    int frame = 4 * t + j - 8;
    if (frame < 0) frame = 0;
    const vint4 v = *(const vint4*)(vcc + frame * 16 + ((i & 3) << 2));
    out4[(size_t)row * C_OUT4 + C_HID4 + i] = v;
}

extern "C" void kernel_launch(void* const* d_in, const int* in_sizes, int n_in,
                              void* d_out, int out_size, void* d_ws, size_t ws_size,
                              hipStream_t stream) {
    const float* hidden = (const float*)d_in[0];
    const float* vcc    = (const float*)d_in[1];
    float* out = (float*)d_out;

    const int nChunks = N_ROWS * CHUNKS_PER_ROW;           // 96768
    const int gridBulk = (nChunks + NBUF - 1) / NBUF;      // 24192 blocks
    bulk_copy_async<<<gridBulk, 256, 0, stream>>>(
        (const vint4*)hidden, (vint4*)out, nChunks);

    const int tailTotal = N_ROWS * TAIL4;                  // 1032192
    tail_group<<<(tailTotal + 255) / 256, 256, 0, stream>>>(
        vcc, (vint4*)out, tailTotal);
}